// VoxelConverter_73435350827170
// MI455X (gfx1250) — compile-verified
//
#include <hip/hip_runtime.h>
#include <cstdint>

// ---------------------------------------------------------------------------
// VoxelConverter for MI455X (gfx1250)
//   pass 1: scatter-max winner pixel per voxel (global_atomic_max_i32)
//   pass 2: per-voxel RGB gather + bilinear feature sampling. Each workgroup's
//           entire 4-tap footprint (12 x 34 pixels x 29 channels, ~46 KB) is
//           DMA'd once into LDS with the Tensor Data Mover. The DMA is issued
//           first and overlapped with the scattered winner/depth/rgb gathers
//           and the non-LDS stores; s_wait_tensorcnt + barrier only before the
//           LDS sampling loop. Output (256 MB streaming) uses NT stores.
// ---------------------------------------------------------------------------

#define IMG_W   512
#define IMG_H   512
#define IMG_HW  (IMG_W * IMG_H)          // 262144
#define VOXEL   64
#define NVOX    (VOXEL * VOXEL * VOXEL)  // 262144
#define NCH     29
#define OUTC    (3 + NCH)                // 32
#define TILE_W  12                       // taps span [8*ix-1, 8*ix+8] -> 10, pad to 12
#define TILE_H  34                       // 4 iy voxels: [8*iy0-1, 8*iy0+32]
#define HALF    1.0f                     // WORLD/2

typedef unsigned int v4u __attribute__((ext_vector_type(4)));
typedef int          v8i __attribute__((ext_vector_type(8)));
typedef int          v4i __attribute__((ext_vector_type(4)));

// ---------------------------------------------------------------- pass 0: init
__global__ __launch_bounds__(256) void vox_init(int* __restrict__ winner) {
  winner[blockIdx.x * 256 + threadIdx.x] = -1;   // empty segment => occ false
}

// ------------------------------------------------------------- pass 1: scatter
__global__ __launch_bounds__(256) void vox_scatter(const float* __restrict__ depth,
                                                   int* __restrict__ winner) {
  const int g = blockIdx.x * 256 + (int)threadIdx.x;   // 0 .. B*HW-1
  const int b = g >> 18;                               // / 262144
  const int p = g & (IMG_HW - 1);
  const float d = depth[g];
  if (!(d > 0.1f && d < 10.0f)) return;

  const int u  = p & (IMG_W - 1);
  const int vp = p >> 9;
  const float X = (u  - 256.0f) * d * (1.0f / 256.0f);  // (u-cx)*d/fx
  const float Y = (vp - 256.0f) * d * (1.0f / 256.0f);

  // (c/HALF + 1)*0.5*64 : pow-2 scalings are exact; (int) == trunc-toward-zero,
  // matching the reference's jnp.trunc quirk (values in (-1,0) land in idx 0).
  const int ix = (int)((X / HALF + 1.0f) * 0.5f * 64.0f);
  const int iy = (int)((Y / HALF + 1.0f) * 0.5f * 64.0f);
  const int iz = (int)((d / HALF + 1.0f) * 0.5f * 64.0f);
  if (ix < 0 || ix >= VOXEL || iy < 0 || iy >= VOXEL || iz < 0 || iz >= VOXEL) return;

  const int lin = (ix << 12) | (iy << 6) | iz;
  atomicMax(&winner[b * NVOX + lin], p);   // segment_max over prio = pixel index
}

// -------------------------------------------------------------- pass 2: gather
__global__ __launch_bounds__(256) void vox_gather(const float* __restrict__ rgb,
                                                  const float* __restrict__ depth,
                                                  const float* __restrict__ feat,
                                                  const int*   __restrict__ winner,
                                                  float*       __restrict__ out) {
  __shared__ __align__(16) float tile[NCH * TILE_H * TILE_W];  // 47328 B

  const int blk = blockIdx.x;
  const int b   = blk >> 10;              // 1024 blocks per batch (NVOX/256)
  const int v0  = (blk & 1023) << 8;      // first voxel of this block
  const int ix  = v0 >> 12;               // constant within block
  const int iy0 = (v0 >> 6) & 63;         // block covers iy0..iy0+3, all iz

  int tx0 = 8 * ix - 1;
  tx0 = tx0 < 0 ? 0 : (tx0 > IMG_W - TILE_W ? IMG_W - TILE_W : tx0);
  int ty0 = 8 * iy0 - 1;
  ty0 = ty0 < 0 ? 0 : (ty0 > IMG_H - TILE_H ? IMG_H - TILE_H : ty0);

  // --- issue TDM DMA first (no wait yet): 3-D tile (x=12, y=34, ch=29) -> LDS
  if (threadIdx.x < 32) {                 // one wave issues the DMA
    const float* src = feat + (size_t)b * NCH * IMG_HW + (size_t)ty0 * IMG_W + tx0;
    const unsigned long long ga = (unsigned long long)(uintptr_t)src;
    const unsigned lds_off = (unsigned)(uintptr_t)(&tile[0]);   // low 32b = LDS offset

    // D# group 0: count=1 | lds_addr | global_addr[56:0] | type=2
    v4u g0 = { 1u,
               lds_off,
               (unsigned)ga,
               (unsigned)((ga >> 32) & 0x01FFFFFFull) | (2u << 30) };
    // D# group 1 (bit-packed per ISA 8.4): wg_mask=0, data_size=2 (4B),
    // tensor_dim0=12, tensor_dim1=34, tile_dim0=12, tile_dim1=34, tile_dim2=29,
    // dim0_stride=512, dim1_stride=262144 (0x40000)
    v8i g1 = { (2 << 16),                  // data_size=4B, no pad/iterate/barrier
               (TILE_W << 16),             // tensor_dim0[15:0] at bit 48
               (TILE_H << 16),             // tensor_dim0 hi=0 | tensor_dim1[15:0]
               (TILE_W << 16),             // tensor_dim1 hi=0 | tile_dim0
               TILE_H | (NCH << 16),       // tile_dim1 | tile_dim2
               IMG_W,                      // tensor_dim0_stride[31:0]
               0,                          // stride0[47:32] | stride1[15:0] (0x40000 -> 0)
               (int)(IMG_HW >> 16) };      // tensor_dim1_stride[47:16] = 4
    v4i g2 = { NCH, 0, 0, 0 };             // tensor_dim2=29, tile_dim3=0
    v4i g3 = { 0, 0, 0, 0 };
#if defined(__clang_major__) && (__clang_major__ >= 23)
    v8i g4 = { 0, 0, 0, 0, 0, 0, 0, 0 };
    __builtin_amdgcn_tensor_load_to_lds(g0, g1, g2, g3, g4, 0);
#else
    __builtin_amdgcn_tensor_load_to_lds(g0, g1, g2, g3, 0);
#endif
  }

  // --- while the DMA runs: scattered gathers + all non-LDS work -------------
  const int v = v0 + (int)threadIdx.x;
  const int w = winner[b * NVOX + v];
  const bool occ = (w >= 0);
  const int wk = occ ? w : 0;
  float* op = out + (size_t)b * OUTC * NVOX + v;

  float wgt[4];
  int   idx[4];

  if (occ) {
    // winner pixel state (scattered loads overlap the tensor DMA)
    const float d  = depth[(size_t)b * IMG_HW + wk];
    const float* rb = rgb + (size_t)b * 3 * IMG_HW + wk;
    const float r0 = rb[0], r1 = rb[IMG_HW], r2 = rb[2 * IMG_HW];

    const int   u  = wk & (IMG_W - 1);
    const int   vp = wk >> 9;
    const float X  = (u  - 256.0f) * d * (1.0f / 256.0f);
    const float Y  = (vp - 256.0f) * d * (1.0f / 256.0f);

    // re-projection: px in [8*ix-0.5, 8*ix+7.5) => taps inside the LDS tile
    const float px = ((X / HALF + 1.0f) * (float)IMG_W - 1.0f) * 0.5f;
    const float py = ((Y / HALF + 1.0f) * (float)IMG_H - 1.0f) * 0.5f;
    const float fx0 = floorf(px), fy0 = floorf(py);
    const float wx1 = px - fx0,   wy1 = py - fy0;
    const int   xi  = (int)fx0,   yi  = (int)fy0;

    #pragma unroll
    for (int t = 0; t < 4; ++t) {
      const int xt = xi + (t & 1);
      const int yt = yi + (t >> 1);
      const float wt = ((t & 1)  ? wx1 : 1.0f - wx1) *
                       ((t >> 1) ? wy1 : 1.0f - wy1);
      const bool inb = (xt >= 0) & (xt < IMG_W) & (yt >= 0) & (yt < IMG_H);
      wgt[t] = inb ? wt : 0.0f;            // out-of-image taps contribute 0
      int lx = xt - tx0; lx = lx < 0 ? 0 : (lx > TILE_W - 1 ? TILE_W - 1 : lx);
      int ly = yt - ty0; ly = ly < 0 ? 0 : (ly > TILE_H - 1 ? TILE_H - 1 : ly);
      idx[t] = ly * TILE_W + lx;           // clamped reads * weight 0 are harmless
    }

    // rgb channels 0..2 out (doesn't need LDS; retire before the barrier)
    __builtin_nontemporal_store(r0, op);
    __builtin_nontemporal_store(r1, op + (size_t)NVOX);
    __builtin_nontemporal_store(r2, op + (size_t)2 * NVOX);
  } else {
    // unoccupied voxel -> all 32 channels zero (also overlaps the DMA)
    #pragma unroll
    for (int c = 0; c < OUTC; ++c)
      __builtin_nontemporal_store(0.0f, op + (size_t)c * NVOX);
  }

  // --- now require the tile: wave 0 drains TENSORcnt, block barriers --------
  if (threadIdx.x < 32)
    __builtin_amdgcn_s_wait_tensorcnt(0);
  __syncthreads();

  if (occ) {
    #pragma unroll 4
    for (int c = 0; c < NCH; ++c) {        // 4 LDS loads + 4 FMAs per channel
      const float* L = &tile[c * (TILE_H * TILE_W)];
      const float s = wgt[0] * L[idx[0]] + wgt[1] * L[idx[1]]
                    + wgt[2] * L[idx[2]] + wgt[3] * L[idx[3]];
      __builtin_nontemporal_store(s, op + (size_t)(3 + c) * NVOX);
    }
  }
}

// ---------------------------------------------------------------------------
extern "C" void kernel_launch(void* const* d_in, const int* in_sizes, int n_in,
                              void* d_out, int out_size, void* d_ws, size_t ws_size,
                              hipStream_t stream) {
  const float* rgb   = (const float*)d_in[0];
  const float* depth = (const float*)d_in[1];
  const float* feat  = (const float*)d_in[2];
  float* out = (float*)d_out;

  const int B = in_sizes[1] / IMG_HW;     // depth has B*H*W elements
  int* winner = (int*)d_ws;               // B * NVOX ints (8 MiB for B=8)

  vox_init   <<< (B * NVOX)   / 256, 256, 0, stream >>> (winner);
  vox_scatter<<< (B * IMG_HW) / 256, 256, 0, stream >>> (depth, winner);
  vox_gather <<< (B * NVOX)   / 256, 256, 0, stream >>> (rgb, depth, feat, winner, out);
}